// MiaoMiaoBlock_65867618452212
// MI455X (gfx1250) — compile-verified
//
#include <hip/hip_runtime.h>
#include <hip/hip_bf16.h>
#include <math.h>

typedef __attribute__((ext_vector_type(2))) float v2f;
typedef __attribute__((ext_vector_type(8))) float v8f;

#define N_RBF 16
#define C 32
#define R_CUT 5.0f
#define EPSV 1e-8f
#define INV_NORM (1.0f / 12.5f)
#define TM 16      // nodes per wave-tile (WMMA M)
#define CS 33      // padded LDS row stride (bank-conflict free)
#define PI_F 3.14159265358979f

// ---------------------------------------------------------------------------
// workspace zeroing (accumulators must be cleared every launch)
// ---------------------------------------------------------------------------
__global__ __launch_bounds__(256) void miao_zero_kernel(float* p, long n) {
    long i = (long)blockIdx.x * blockDim.x + threadIdx.x;
    long stride = (long)gridDim.x * blockDim.x;
    for (; i < n; i += stride) p[i] = 0.0f;
}

// ---------------------------------------------------------------------------
// edge kernel: rbf filters + moment tensor products + atomic scatter
// acc0:[N,32] acc1:[N,32,3] acc2:[N,32,3,3] (raw sums; /12.5 applied later)
// ---------------------------------------------------------------------------
__global__ __launch_bounds__(256) void miao_edge_kernel(
    const float* __restrict__ t0, const float* __restrict__ t1,
    const float* __restrict__ t2, const float* __restrict__ rij,
    const float* __restrict__ Wrbf, const int* __restrict__ idx_i,
    const int* __restrict__ idx_j, float* __restrict__ acc0,
    float* __restrict__ acc1, float* __restrict__ acc2, int E) {
    __shared__ float sW[3 * N_RBF * C];  // 6 KB
    for (int i = threadIdx.x; i < 3 * N_RBF * C; i += 256) sW[i] = Wrbf[i];
    __syncthreads();

    int e = blockIdx.x * 256 + threadIdx.x;
    if (e >= E) return;

    float rx = rij[e * 3 + 0], ry = rij[e * 3 + 1], rz = rij[e * 3 + 2];
    float d = sqrtf(rx * rx + ry * ry + rz * rz + EPSV);
    float inv_d = 1.0f / d;
    float u[3] = {rx * inv_d, ry * inv_d, rz * inv_d};

    // Gaussian RBF with cosine cutoff
    float dc = fminf(d, R_CUT);
    float fcut = 0.5f * (__cosf(PI_F * dc / R_CUT) + 1.0f);
    const float mu_step = R_CUT / (float)(N_RBF - 1);
    const float inv_delta = (float)N_RBF / R_CUT;
    float rbf[N_RBF];
#pragma unroll
    for (int k = 0; k < N_RBF; k++) {
        float t = (d - mu_step * (float)k) * inv_delta;
        rbf[k] = __expf(-t * t) * fcut;
    }

    int i_n = idx_i[e];
    int j_n = idx_j[e];
    const float* h0p = t0 + (long)j_n * C;
    const float* h1p = t1 + (long)j_n * C * 3;
    const float* h2p = t2 + (long)j_n * C * 9;
    float* a0p = acc0 + (long)i_n * C;
    float* a1p = acc1 + (long)i_n * C * 3;
    float* a2p = acc2 + (long)i_n * C * 9;

    for (int c = 0; c < C; c++) {
        float f0 = 0.f, f1 = 0.f, f2 = 0.f;
#pragma unroll
        for (int k = 0; k < N_RBF; k++) {
            float r = rbf[k];
            f0 += r * sW[0 * N_RBF * C + k * C + c];
            f1 += r * sW[1 * N_RBF * C + k * C + c];
            f2 += r * sW[2 * N_RBF * C + k * C + c];
        }
        float h0 = h0p[c];
        float h1[3], h2[9];
#pragma unroll
        for (int p = 0; p < 3; p++) h1[p] = h1p[c * 3 + p];
#pragma unroll
        for (int p = 0; p < 9; p++) h2[p] = h2p[c * 9 + p];

        // a1 = h1.u ; v = h2.u ; a2 = u.h2.u = v.u  (since m2 = u u^T)
        float a1 = h1[0] * u[0] + h1[1] * u[1] + h1[2] * u[2];
        float v[3];
#pragma unroll
        for (int p = 0; p < 3; p++)
            v[p] = h2[p * 3 + 0] * u[0] + h2[p * 3 + 1] * u[1] + h2[p * 3 + 2] * u[2];
        float a2 = v[0] * u[0] + v[1] * u[1] + v[2] * u[2];

        // msg0
        unsafeAtomicAdd(a0p + c, f0 * h0 + f1 * a1 + f2 * a2);
        // msg1[p] = f1*h0*u_p + f0*h1_p + f2*a1*u_p + f1*v_p
        float c01 = f1 * h0 + f2 * a1;
#pragma unroll
        for (int p = 0; p < 3; p++)
            unsafeAtomicAdd(a1p + c * 3 + p, c01 * u[p] + f0 * h1[p] + f1 * v[p]);
        // msg2[p][q] = f2*h0*u_p*u_q + f1*h1_p*u_q + f0*h2_pq + f2*v_p*u_q
#pragma unroll
        for (int p = 0; p < 3; p++) {
            float cp = f2 * h0 * u[p] + f1 * h1[p] + f2 * v[p];
#pragma unroll
            for (int q = 0; q < 3; q++)
                unsafeAtomicAdd(a2p + c * 9 + p * 3 + q, cp * u[q] + f0 * h2[p * 3 + q]);
        }
    }
}

// ---------------------------------------------------------------------------
// fp32 WMMA helpers: 16x(K=32)x16 GEMM tile via 8 chained V_WMMA_F32_16X16X4_F32
// A: LDS, row stride CS (16 rows x 32 cols). B: LDS row-major [32][32].
// ---------------------------------------------------------------------------
__device__ inline v8f miao_gemm_tile(const float* A, const float* B, int n0,
                                     float bias_n, int lane) {
    int m = lane & 15;
    int kh = (lane < 16) ? 0 : 2;  // K split across lane halves (ISA A/B layout)
    int n = (lane & 15) + n0;
    v8f acc;
#pragma unroll
    for (int i = 0; i < 8; i++) acc[i] = bias_n;
#pragma unroll
    for (int kb = 0; kb < 8; kb++) {
        v2f a, b;
        a.x = A[m * CS + 4 * kb + kh];
        a.y = A[m * CS + 4 * kb + kh + 1];
        b.x = B[(4 * kb + kh) * C + n];
        b.y = B[(4 * kb + kh + 1) * C + n];
        acc = __builtin_amdgcn_wmma_f32_16x16x4_f32(false, a, false, b, (short)0,
                                                    acc, false, false);
    }
    return acc;
}

__device__ inline void miao_store_tile(float* dst, v8f acc, int n0, int lane) {
    int n = (lane & 15) + n0;
    int mb = (lane < 16) ? 0 : 8;  // ISA C/D layout: M = vgpr + 8*(lane>=16)
#pragma unroll
    for (int r = 0; r < 8; r++) dst[(mb + r) * CS + n] = acc[r];
}

__device__ inline float miao_silu(float x) { return x / (1.0f + __expf(-x)); }

// ---------------------------------------------------------------------------
// node kernel: one wave per 16 nodes.  s = acc/12.5 + t ; y = s @ W_si ;
// gates via W_nl ; out = z + s.   All 6 GEMMs use fp32 WMMA.
// ---------------------------------------------------------------------------
__global__ __launch_bounds__(32) void miao_node_kernel(
    const float* __restrict__ t0, const float* __restrict__ t1,
    const float* __restrict__ t2, const float* __restrict__ Wsi,
    const float* __restrict__ bsi0, const float* __restrict__ Wnl,
    const float* __restrict__ bnl, const float* __restrict__ acc0,
    const float* __restrict__ acc1, const float* __restrict__ acc2,
    float* __restrict__ out0, float* __restrict__ out1,
    float* __restrict__ out2, int N) {
    __shared__ float sWsi[3 * C * C];   // 12 KB
    __shared__ float sWnl[3 * C * C];   // 12 KB
    __shared__ float sBsi0[C];
    __shared__ float sBnl[3 * C];
    __shared__ float sA[TM * CS];       // A-matrix staging
    __shared__ float sY[9][TM * CS];    // y components
    __shared__ float sG[TM * CS];       // gate

    int lane = threadIdx.x;
    for (int i = lane; i < 3 * C * C; i += 32) { sWsi[i] = Wsi[i]; sWnl[i] = Wnl[i]; }
    if (lane < C) sBsi0[lane] = bsi0[lane];
    for (int i = lane; i < 3 * C; i += 32) sBnl[i] = bnl[i];
    __syncthreads();

    int node0 = blockIdx.x * TM;
    int rows = N - node0;
    if (rows > TM) rows = TM;

    // ================= way 0 =================
    for (int idx = lane; idx < TM * C; idx += 32) {
        int m = idx >> 5, k = idx & 31;
        float v = 0.f;
        if (m < rows) {
            long o = (long)(node0 + m) * C + k;
            v = acc0[o] * INV_NORM + t0[o];
        }
        sA[m * CS + k] = v;
    }
    __syncthreads();
#pragma unroll
    for (int t = 0; t < 2; t++) {
        v8f d = miao_gemm_tile(sA, sWsi, t * 16, sBsi0[(lane & 15) + t * 16], lane);
        miao_store_tile(sY[0], d, t * 16, lane);  // y0
    }
    __syncthreads();
#pragma unroll
    for (int t = 0; t < 2; t++) {
        v8f g = miao_gemm_tile(sY[0], sWnl, t * 16, sBnl[(lane & 15) + t * 16], lane);
        int n = (lane & 15) + t * 16;
        int mb = (lane < 16) ? 0 : 8;
#pragma unroll
        for (int r = 0; r < 8; r++) {
            int m = mb + r;
            if (m < rows) {
                long o = (long)(node0 + m) * C + n;
                float s0 = acc0[o] * INV_NORM + t0[o];
                out0[o] = miao_silu(g[r]) + s0;
            }
        }
    }
    __syncthreads();

    // ================= way 1 =================
    for (int r = 0; r < 3; r++) {
        for (int idx = lane; idx < TM * C; idx += 32) {
            int m = idx >> 5, k = idx & 31;
            float v = 0.f;
            if (m < rows) {
                long n = node0 + m;
                v = acc1[(n * C + k) * 3 + r] * INV_NORM + t1[n * C * 3 + k * 3 + r];
            }
            sA[m * CS + k] = v;
        }
        __syncthreads();
#pragma unroll
        for (int t = 0; t < 2; t++) {
            v8f d = miao_gemm_tile(sA, sWsi + C * C, t * 16, 0.0f, lane);
            miao_store_tile(sY[r], d, t * 16, lane);
        }
        __syncthreads();
    }
    for (int idx = lane; idx < TM * C; idx += 32) {
        int m = idx >> 5, k = idx & 31;
        int o = m * CS + k;
        float a = sY[0][o], b = sY[1][o], c = sY[2][o];
        sA[o] = sqrtf(a * a + b * b + c * c + EPSV);  // n1
    }
    __syncthreads();
#pragma unroll
    for (int t = 0; t < 2; t++) {
        v8f g = miao_gemm_tile(sA, sWnl + C * C, t * 16,
                               sBnl[C + (lane & 15) + t * 16], lane);
        int n = (lane & 15) + t * 16;
        int mb = (lane < 16) ? 0 : 8;
#pragma unroll
        for (int r = 0; r < 8; r++) sG[(mb + r) * CS + n] = miao_silu(g[r]);
    }
    __syncthreads();
    for (int idx = lane; idx < TM * C; idx += 32) {
        int m = idx >> 5, k = idx & 31;
        if (m < rows) {
            long n = node0 + m;
            float gate = sG[m * CS + k];
#pragma unroll
            for (int r = 0; r < 3; r++) {
                float s1 = acc1[(n * C + k) * 3 + r] * INV_NORM + t1[n * C * 3 + k * 3 + r];
                out1[n * C * 3 + k * 3 + r] = sY[r][m * CS + k] * gate + s1;
            }
        }
    }
    __syncthreads();

    // ================= way 2 =================
    for (int r = 0; r < 9; r++) {
        for (int idx = lane; idx < TM * C; idx += 32) {
            int m = idx >> 5, k = idx & 31;
            float v = 0.f;
            if (m < rows) {
                long n = node0 + m;
                v = acc2[(n * C + k) * 9 + r] * INV_NORM + t2[n * C * 9 + k * 9 + r];
            }
            sA[m * CS + k] = v;
        }
        __syncthreads();
#pragma unroll
        for (int t = 0; t < 2; t++) {
            v8f d = miao_gemm_tile(sA, sWsi + 2 * C * C, t * 16, 0.0f, lane);
            miao_store_tile(sY[r], d, t * 16, lane);
        }
        __syncthreads();
    }
    for (int idx = lane; idx < TM * C; idx += 32) {
        int m = idx >> 5, k = idx & 31;
        int o = m * CS + k;
        float acc = EPSV;
#pragma unroll
        for (int r = 0; r < 9; r++) { float y = sY[r][o]; acc += y * y; }
        sA[o] = sqrtf(acc);  // n2
    }
    __syncthreads();
#pragma unroll
    for (int t = 0; t < 2; t++) {
        v8f g = miao_gemm_tile(sA, sWnl + 2 * C * C, t * 16,
                               sBnl[2 * C + (lane & 15) + t * 16], lane);
        int n = (lane & 15) + t * 16;
        int mb = (lane < 16) ? 0 : 8;
#pragma unroll
        for (int r = 0; r < 8; r++) sG[(mb + r) * CS + n] = miao_silu(g[r]);
    }
    __syncthreads();
    for (int idx = lane; idx < TM * C; idx += 32) {
        int m = idx >> 5, k = idx & 31;
        if (m < rows) {
            long n = node0 + m;
            float gate = sG[m * CS + k];
#pragma unroll
            for (int r = 0; r < 9; r++) {
                float s2 = acc2[(n * C + k) * 9 + r] * INV_NORM + t2[n * C * 9 + k * 9 + r];
                out2[n * C * 9 + k * 9 + r] = sY[r][m * CS + k] * gate + s2;
            }
        }
    }
}

// ---------------------------------------------------------------------------
extern "C" void kernel_launch(void* const* d_in, const int* in_sizes, int n_in,
                              void* d_out, int out_size, void* d_ws, size_t ws_size,
                              hipStream_t stream) {
    const float* t0   = (const float*)d_in[0];
    const float* t1   = (const float*)d_in[1];
    const float* t2   = (const float*)d_in[2];
    const float* rij  = (const float*)d_in[3];
    const float* Wrbf = (const float*)d_in[4];
    const float* Wsi  = (const float*)d_in[5];
    const float* bsi0 = (const float*)d_in[6];
    const float* Wnl  = (const float*)d_in[7];
    const float* bnl  = (const float*)d_in[8];
    const int* idx_i  = (const int*)d_in[9];
    const int* idx_j  = (const int*)d_in[10];

    const int N = in_sizes[0] / C;   // 20000
    const int E = in_sizes[9];       // 250000

    float* acc0 = (float*)d_ws;                 // [N,32]
    float* acc1 = acc0 + (long)N * C;           // [N,32,3]
    float* acc2 = acc1 + (long)N * C * 3;       // [N,32,3,3]
    float* out0 = (float*)d_out;
    float* out1 = out0 + (long)N * C;
    float* out2 = out1 + (long)N * C * 3;

    long acc_total = (long)N * (C * 13);
    miao_zero_kernel<<<512, 256, 0, stream>>>((float*)d_ws, acc_total);

    int eblocks = (E + 255) / 256;
    miao_edge_kernel<<<eblocks, 256, 0, stream>>>(t0, t1, t2, rij, Wrbf, idx_i,
                                                  idx_j, acc0, acc1, acc2, E);

    int nblocks = (N + TM - 1) / TM;
    miao_node_kernel<<<nblocks, 32, 0, stream>>>(t0, t1, t2, Wsi, bsi0, Wnl, bnl,
                                                 acc0, acc1, acc2, out0, out1,
                                                 out2, N);
}